// NCELoss_87591563034662
// MI455X (gfx1250) — compile-verified
//
#include <hip/hip_runtime.h>
#include <hip/hip_bf16.h>

// ---------------------------------------------------------------------------
// NCE-style loss, B=4096 rows x N=32768 cols, f32.
// HBM-bound: 1 GiB stream @ 23.3 TB/s => ~46us floor. Strategy:
//   * uniform (scalar-branch) loop, unroll x4 => 8 outstanding 128-bit NT
//     loads per wave per batch (deep MLP for the HBM stream)
//   * v_exp_f32 / v_rcp_f32 for sigmoid+exp (trans units are not the limit)
//   * 4 independent accumulator chains to hide v_exp/v_add latency
//   * cross-lane reduction via V_WMMA_F32_16X16X4_F32 (exact f32 reduce):
//       A[m][K0..1] = v(lane m), A[m][K2..3] = v(lane m+16)
//       B = splat(0.5)  (layout-agnostic: every entry identical)
//       => D[m][n] = v(m) + v(m+16); sum own 8 acc VGPRs = half-wave sum,
//          one shfl_xor(16) = full 32-lane sum.
// ---------------------------------------------------------------------------

typedef __attribute__((ext_vector_type(2))) float v2f;
typedef __attribute__((ext_vector_type(4))) float v4f;
typedef __attribute__((ext_vector_type(8))) float v8f;

#define NCE_INV_T 14.2857142857142857f   // 1 / 0.07
#define NCE_EPS   1e-12f
#define BLOCK     256

// Exact 32-lane f32 sum using one WMMA; EXEC must be all-1s at the call site.
__device__ __forceinline__ float wave_sum32_wmma(float v) {
    v2f a; a.x = v;    a.y = v;      // A 16x4: lanes0-15 -> K0,K1 ; lanes16-31 -> K2,K3
    v2f b; b.x = 0.5f; b.y = 0.5f;   // B 4x16: all entries 0.5 (layout agnostic)
    v8f c = {};
    c = __builtin_amdgcn_wmma_f32_16x16x4_f32(false, a, false, b, (short)0, c,
                                              false, false);
    // lane L<16 holds D[0..7][L] (rows 0-7); lane L>=16 holds D[8..15][L-16]
    float s = c[0] + c[1] + c[2] + c[3] + c[4] + c[5] + c[6] + c[7];
    s += __shfl_xor(s, 16, 32);      // combine the two row-halves
    return s;                        // every lane: sum of all 32 inputs
}

__global__ __launch_bounds__(BLOCK) void nce_row_kernel(
    const float* __restrict__ scores,
    const float* __restrict__ labels,
    float* __restrict__ row_loss,
    int iters) {                     // = N / 4 / BLOCK (uniform, 32)
    const int row = blockIdx.x;
    const int tid = threadIdx.x;

    const size_t N = (size_t)iters * 4u * BLOCK;
    const v4f* __restrict__ s4 =
        reinterpret_cast<const v4f*>(scores + (size_t)row * N);
    const v4f* __restrict__ l4 =
        reinterpret_cast<const v4f*>(labels + (size_t)row * N);

    float pn0 = 0.f, pn1 = 0.f, pn2 = 0.f, pn3 = 0.f;
    float pd0 = 0.f, pd1 = 0.f, pd2 = 0.f, pd3 = 0.f;

    // Scalar loop counter -> s_cbranch loop (no per-lane exec masking).
    // 4x batch: issue all 8 b128 NT loads before consuming any.
    for (int j = 0; j < iters; j += 4) {
        v4f x[4], y[4];
#pragma unroll
        for (int u = 0; u < 4; ++u) {
            const int i = tid + (j + u) * BLOCK;
            x[u] = __builtin_nontemporal_load(s4 + i);
            y[u] = __builtin_nontemporal_load(l4 + i);
        }
        float en[4], ed[4];
#pragma unroll
        for (int u = 0; u < 4; ++u) {
            float n = 0.f, d = 0.f;
#pragma unroll
            for (int k = 0; k < 4; ++k) {
                float sig = __builtin_amdgcn_rcpf(1.0f + __expf(-x[u][k]));
                float e   = __expf(sig * NCE_INV_T);   // v_exp_f32
                d += e;
                n = fmaf(e, y[u][k], n);
            }
            en[u] = n; ed[u] = d;
        }
        // 4 independent accumulator chains
        pn0 += en[0]; pn1 += en[1]; pn2 += en[2]; pn3 += en[3];
        pd0 += ed[0]; pd1 += ed[1]; pd2 += ed[2]; pd3 += ed[3];
    }

    float pn = (pn0 + pn1) + (pn2 + pn3);
    float pd = (pd0 + pd1) + (pd2 + pd3);

    // Wave-level exact reduction via WMMA (uniform flow -> EXEC all 1s here)
    pn = wave_sum32_wmma(pn);
    pd = wave_sum32_wmma(pd);

    __shared__ float sn[8], sd[8];
    const int wave = tid >> 5;
    if ((tid & 31) == 0) { sn[wave] = pn; sd[wave] = pd; }
    __syncthreads();

    if (tid == 0) {
        float num = 0.0f, den = 0.0f;
#pragma unroll
        for (int w = 0; w < 8; ++w) { num += sn[w]; den += sd[w]; }
        row_loss[row] = -__logf(num / (den + NCE_EPS) + NCE_EPS);
    }
}

__global__ __launch_bounds__(BLOCK) void nce_mean_kernel(
    const float* __restrict__ row_loss,
    float* __restrict__ out,
    int B) {
    const int tid = threadIdx.x;
    float p = 0.0f;
    for (int j = 0; j < B / BLOCK; ++j)          // uniform scalar loop
        p += row_loss[tid + j * BLOCK];

    p = wave_sum32_wmma(p);

    __shared__ float sw[8];
    if ((tid & 31) == 0) sw[tid >> 5] = p;
    __syncthreads();

    if (tid == 0) {
        float s = 0.0f;
#pragma unroll
        for (int w = 0; w < 8; ++w) s += sw[w];
        out[0] = s / (float)B;
    }
}

extern "C" void kernel_launch(void* const* d_in, const int* in_sizes, int n_in,
                              void* d_out, int out_size, void* d_ws, size_t ws_size,
                              hipStream_t stream) {
    const float* scores = (const float*)d_in[0];
    const float* labels = (const float*)d_in[1];
    float* out = (float*)d_out;

    const int N = 32768;
    const int B = in_sizes[0] / N;               // 4096
    const int iters = N / 4 / BLOCK;             // 32

    float* row_loss = (float*)d_ws;              // B floats of scratch (16 KiB)

    nce_row_kernel<<<B, BLOCK, 0, stream>>>(scores, labels, row_loss, iters);
    nce_mean_kernel<<<1, BLOCK, 0, stream>>>(row_loss, out, B);
}